// ODEfunc2_40956808135041
// MI455X (gfx1250) — compile-verified
//
#include <hip/hip_runtime.h>

// CDNA5 / gfx1250: wave32, WMMA f32 16x16x4 for fp32 GEMM.
typedef __attribute__((ext_vector_type(2))) float v2f;
typedef __attribute__((ext_vector_type(8))) float v8f;

#define DIM 128
#define LDS_STRIDE 130  // pad: keeps 8B alignment (even), spreads banks

// ---------------------------------------------------------------- zero fill
__global__ __launch_bounds__(256) void zero4_kernel(float4* __restrict__ p, int n4) {
  int i = blockIdx.x * 256 + threadIdx.x;
  if (i < n4) p[i] = make_float4(0.f, 0.f, 0.f, 0.f);
}

// ---------------------------------------------------------------- GEMM
// S[N,128] = concat([t, X], 1) @ Wfull[129,128] + bias
//          = X @ Wfull[1:,:] + (bias + t * Wfull[0,:])
// Block: 256 thr = 8 waves; each wave -> 16 rows x 128 cols.
// Per wave: 8 accumulators (16x16), K looped in chunks of 4 -> 256 WMMAs.
__global__ __launch_bounds__(256) void gemm_wmma_kernel(
    const float* __restrict__ X, const float* __restrict__ Wfull,
    const float* __restrict__ bias, const float* __restrict__ tptr,
    float* __restrict__ S, int N) {
  __shared__ float WT[DIM * LDS_STRIDE];  // W transposed: WT[n][k] = W[k+1][n]

  const int tid = threadIdx.x;
  const float* Wp = Wfull + DIM;  // skip t-row
  // Cooperative staging: coalesced global reads, transposed LDS writes.
  for (int j = 0; j < 64; ++j) {
    int i = tid + 256 * j;         // 0 .. 16383
    int k = i >> 7;                // 0 .. 127
    int n = i & 127;               // 0 .. 127
    WT[n * LDS_STRIDE + k] = Wp[k * DIM + n];
  }
  __syncthreads();

  const int wave = tid >> 5;
  const int lane = tid & 31;
  const int m  = lane & 15;   // M (A) / N (B,C) sub-index
  const int hh = lane >> 4;   // half-wave select (K pair / M+8)
  const int rowbase = blockIdx.x * 128 + wave * 16;
  const float tval = tptr[0];

  v8f acc[8] = {};

  // A-frag source row (clamped for tail block; stores are guarded).
  int arow = rowbase + m;
  if (arow >= N) arow = N - 1;
  const float* xrow = X + (size_t)arow * DIM;

  for (int kc = 0; kc < 32; ++kc) {
    const int k0 = kc * 4;
    // A 16x4 layout: lane m holds A[m][k0+2*hh], A[m][k0+2*hh+1]
    v2f a = *(const v2f*)(xrow + k0 + 2 * hh);
#pragma unroll
    for (int nt = 0; nt < 8; ++nt) {
      // B 4x16 layout: lane holds B[k0+2*hh][n], B[k0+2*hh+1][n], n = nt*16+m
      v2f b = *(const v2f*)&WT[(nt * 16 + m) * LDS_STRIDE + k0 + 2 * hh];
      acc[nt] = __builtin_amdgcn_wmma_f32_16x16x4_f32(
          false, a, false, b, (short)0, acc[nt], false, false);
    }
  }

  // C/D 16x16 layout: col = nt*16 + m, row = rowbase + r + 8*hh (VGPR r)
#pragma unroll
  for (int nt = 0; nt < 8; ++nt) {
    const int col = nt * 16 + m;
    const float be = bias[col] + tval * Wfull[col];  // fold t-row + bias
#pragma unroll
    for (int r = 0; r < 8; ++r) {
      const int row = rowbase + r + 8 * hh;
      if (row < N) S[(size_t)row * DIM + col] = acc[nt][r] + be;
    }
  }
}

// ---------------------------------------------------------------- scatter-add
// One wave per edge: gather one 512B row of S (32 lanes x float4), scale by
// edge weight, hardware-f32-atomic-add into H[tgt]. All L2-resident.
__global__ __launch_bounds__(256) void scatter_kernel(
    const float* __restrict__ S, const int* __restrict__ src,
    const int* __restrict__ tgt, const float* __restrict__ ew,
    float* __restrict__ H, int E) {
  const int e = blockIdx.x * 8 + (threadIdx.x >> 5);
  const int lane = threadIdx.x & 31;
  if (e >= E) return;
  const int si = src[e];
  const int ti = tgt[e];
  const float w = ew[e];
  float4 v = ((const float4*)(S + (size_t)si * DIM))[lane];
  float* hp = H + (size_t)ti * DIM + lane * 4;
  unsafeAtomicAdd(hp + 0, v.x * w);
  unsafeAtomicAdd(hp + 1, v.y * w);
  unsafeAtomicAdd(hp + 2, v.z * w);
  unsafeAtomicAdd(hp + 3, v.w * w);
}

// ---------------------------------------------------------------- ReLU + GN
// 32 groups x 4 channels: each lane owns exactly one group as a float4 ->
// fully lane-local statistics, no cross-lane ops. One wave per row.
__global__ __launch_bounds__(256) void relu_gn_kernel(
    const float* __restrict__ H, const float* __restrict__ gamma,
    const float* __restrict__ beta, float* __restrict__ O, int N) {
  const int row = blockIdx.x * 8 + (threadIdx.x >> 5);
  const int lane = threadIdx.x & 31;
  if (row >= N) return;
  float4 v = ((const float4*)(H + (size_t)row * DIM))[lane];
  v.x = fmaxf(v.x, 0.f);
  v.y = fmaxf(v.y, 0.f);
  v.z = fmaxf(v.z, 0.f);
  v.w = fmaxf(v.w, 0.f);
  const float mu = (v.x + v.y + v.z + v.w) * 0.25f;
  const float dx = v.x - mu, dy = v.y - mu, dz = v.z - mu, dw = v.w - mu;
  const float var = (dx * dx + dy * dy + dz * dz + dw * dw) * 0.25f;
  const float rs = rsqrtf(var + 1e-5f);
  const float4 g = ((const float4*)gamma)[lane];
  const float4 b = ((const float4*)beta)[lane];
  float4 o;
  o.x = dx * rs * g.x + b.x;
  o.y = dy * rs * g.y + b.y;
  o.z = dz * rs * g.z + b.z;
  o.w = dw * rs * g.w + b.w;
  ((float4*)(O + (size_t)row * DIM))[lane] = o;
}

// ---------------------------------------------------------------- launch
extern "C" void kernel_launch(void* const* d_in, const int* in_sizes, int n_in,
                              void* d_out, int out_size, void* d_ws, size_t ws_size,
                              hipStream_t stream) {
  const float* t   = (const float*)d_in[0];
  const float* x   = (const float*)d_in[1];
  const int*   src = (const int*)d_in[2];
  const int*   tgt = (const int*)d_in[3];
  const float* ew  = (const float*)d_in[4];
  const float* W1  = (const float*)d_in[5];
  const float* b1  = (const float*)d_in[6];
  const float* W2  = (const float*)d_in[7];
  const float* b2  = (const float*)d_in[8];
  const float* g1  = (const float*)d_in[9];
  const float* be1 = (const float*)d_in[10];
  const float* g2  = (const float*)d_in[11];
  const float* be2 = (const float*)d_in[12];

  const int N = in_sizes[1] / DIM;
  const int E = in_sizes[2];

  float* s_buf = (float*)d_ws;                       // [N,128] projected support
  float* h_buf = s_buf + (size_t)N * DIM;            // [N,128] aggregation accum
  float* out   = (float*)d_out;

  const dim3 blk(256);
  const int gemmGrid = (N + 127) / 128;
  const int edgeGrid = (E + 7) / 8;
  const int rowGrid  = (N + 7) / 8;
  const int n4       = N * DIM / 4;
  const int zeroGrid = (n4 + 255) / 256;

  // ---- layer 1
  zero4_kernel<<<zeroGrid, blk, 0, stream>>>((float4*)h_buf, n4);
  gemm_wmma_kernel<<<gemmGrid, blk, 0, stream>>>(x, W1, b1, t, s_buf, N);
  scatter_kernel<<<edgeGrid, blk, 0, stream>>>(s_buf, src, tgt, ew, h_buf, E);
  relu_gn_kernel<<<rowGrid, blk, 0, stream>>>(h_buf, g1, be1, h_buf, N);

  // ---- layer 2 (accumulate straight into d_out)
  zero4_kernel<<<zeroGrid, blk, 0, stream>>>((float4*)out, n4);
  gemm_wmma_kernel<<<gemmGrid, blk, 0, stream>>>(h_buf, W2, b2, t, s_buf, N);
  scatter_kernel<<<edgeGrid, blk, 0, stream>>>(s_buf, src, tgt, ew, out, E);
  relu_gn_kernel<<<rowGrid, blk, 0, stream>>>(out, g2, be2, out, N);
}